// SSDBlock_15075335209599
// MI455X (gfx1250) — compile-verified
//
#include <hip/hip_runtime.h>
#include <hip/hip_bf16.h>

// ---------------------------------------------------------------------------
// Mamba-2 / SSD block for MI455X (gfx1250, wave32).
//  * All matmuls -> V_WMMA_F32_16X16X4_F32 (fp32-exact, matches reference).
//  * All GEMM / scan tile staging -> Tensor Data Mover (tensor_load_to_lds),
//    double-buffered in LDS, synchronized with s_wait_tensorcnt.
// ---------------------------------------------------------------------------

#define D_MODEL   1024
#define EVENT_DIM 16
#define D_AUG     (D_MODEL + EVENT_DIM)   // 1040
#define D_INNER   2048
#define N_HEADS   32
#define HEAD_DIM  64
#define D_STATE   64
#define D_CONV    4
#define SEQ_L     4096
#define BATCH     4
#define ROWS      (BATCH * SEQ_L)         // 16384
#define CHUNK     16
#define NCHUNK    (SEQ_L / CHUNK)         // 256
#define LN_EPS    1e-5f

typedef float v2f __attribute__((ext_vector_type(2)));
typedef float v8f __attribute__((ext_vector_type(8)));
typedef unsigned int u32x4 __attribute__((ext_vector_type(4)));
typedef int i32x4 __attribute__((ext_vector_type(4)));
typedef int i32x8 __attribute__((ext_vector_type(8)));

// D = A(16x4) * B(4x16) + C -- fp32 WMMA, wave32. Layout per ISA 7.12.2.
static __device__ inline v8f wmma4(v2f a, v2f b, v8f c) {
    return __builtin_amdgcn_wmma_f32_16x16x4_f32(
        false, a, false, b, (short)0, c, false, false);
}

static __device__ inline float siluf(float v)     { return v / (1.0f + __expf(-v)); }
static __device__ inline float softplusf(float v) { return (v > 20.0f) ? v : log1pf(__expf(v)); }

// Low 32 bits of a generic pointer into LDS == LDS byte offset (ISA 10.2).
static __device__ inline unsigned lds_addr_of(const void* p) {
    return (unsigned)(unsigned long long)(uintptr_t)p;
}

// ---------------------------------------------------------------------------
// TDM: async-load a 2-D tile (tile1 rows x tile0 contiguous elems, fp32) from
// global (row stride = stride0 elems) into LDS at ldsAddr. D# per ISA 8.3/8.4.
// Issue from one wave only; completion via s_wait_tensorcnt.
// 6-arg builtin form: (g0 u32x4, g1 i32x8, g2 i32x4, g3 i32x4, i32x8, cpol).
// ---------------------------------------------------------------------------
static __device__ inline void tdm_load_2d(unsigned ldsAddr, const float* gptr,
                                          unsigned tile0, unsigned tile1,
                                          unsigned stride0) {
    const unsigned long long ga = (unsigned long long)(uintptr_t)gptr;
    u32x4 g0;
    g0.x = 1u;                                        // count=1 (valid), user mode
    g0.y = ldsAddr;                                   // lds_addr [63:32]
    g0.z = (unsigned)ga;                              // global_addr[31:0]
    g0.w = ((unsigned)(ga >> 32) & 0x01FFFFFFu) | (2u << 30);  // addr[56:32] | type=2
    i32x8 g1;
    g1[0] = 0x00020000;                               // data_size=2 (4B), mask=0
    g1[1] = (int)((tile0 & 0xFFFFu) << 16);           // tensor_dim0[15:0] (=tile0)
    g1[2] = (int)(((tile0 >> 16) & 0xFFFFu) |
                  ((tile1 & 0xFFFFu) << 16));         // td0 hi | tensor_dim1 lo (=tile1)
    g1[3] = (int)(((tile1 >> 16) & 0xFFFFu) |
                  ((tile0 & 0xFFFFu) << 16));         // td1 hi | tile_dim0
    g1[4] = (int)(tile1 & 0xFFFFu);                   // tile_dim1 (tile_dim2=0 -> 2D)
    g1[5] = (int)stride0;                             // tensor_dim0_stride[31:0]
    g1[6] = 0;                                        // stride hi | dim1_stride lo
    g1[7] = 0;
    const i32x4 gz4 = {0, 0, 0, 0};
    const i32x8 gz8 = {0, 0, 0, 0, 0, 0, 0, 0};
    __builtin_amdgcn_tensor_load_to_lds(g0, g1, gz4, gz4, gz8, 0);
}

// ---------------------------------------------------------------------------
// Tiled WMMA GEMM with TDM double buffering.
// C[MxN] = A[MxK] @ B[KxN], row-major, M%128==0, N%128==0, K%16==0.
// Block: 256 thr (8 waves), 128x128 tile; wave (mw 0..3, nw 0..1) owns 32x64.
// ---------------------------------------------------------------------------
#define BM 128
#define BN 128
#define BK 16

__global__ __launch_bounds__(256)
void gemm_tdm_wmma(const float* __restrict__ A, const float* __restrict__ B,
                   float* __restrict__ C, int M, int N, int K) {
    __shared__ float sA[2][BM][BK];
    __shared__ float sB[2][BK][BN];

    const int tid  = threadIdx.x;
    const int wave = tid >> 5;
    const int lane = tid & 31;
    const int half = lane >> 4;
    const int lr   = lane & 15;
    const int mw   = wave >> 1;          // 0..3 -> 32-row slab
    const int nw   = wave & 1;           // 0..1 -> 64-col slab
    const int m0   = blockIdx.y * BM;
    const int n0   = blockIdx.x * BN;
    const int S    = K / BK;

    v8f acc[2][4];
#pragma unroll
    for (int mt = 0; mt < 2; ++mt)
#pragma unroll
        for (int nt = 0; nt < 4; ++nt) acc[mt][nt] = (v8f){};

    if (tid < 32) {                       // prologue: stage 0 -> buf 0
        tdm_load_2d(lds_addr_of(&sA[0][0][0]), A + (long)m0 * K, BK, BM, (unsigned)K);
        tdm_load_2d(lds_addr_of(&sB[0][0][0]), B + (long)n0,     BN, BK, (unsigned)N);
    }

    for (int s = 0; s < S; ++s) {
        const int cur = s & 1;
        if (tid < 32) {
            if (s + 1 < S) {
                const int nxt = (s + 1) & 1;
                tdm_load_2d(lds_addr_of(&sA[nxt][0][0]),
                            A + (long)m0 * K + (s + 1) * BK, BK, BM, (unsigned)K);
                tdm_load_2d(lds_addr_of(&sB[nxt][0][0]),
                            B + (long)(s + 1) * BK * N + n0, BN, BK, (unsigned)N);
                __builtin_amdgcn_s_wait_tensorcnt(2);   // stage s complete
            } else {
                __builtin_amdgcn_s_wait_tensorcnt(0);
            }
        }
        __syncthreads();

#pragma unroll
        for (int k0 = 0; k0 < BK; k0 += 4) {
            const int ka = k0 + 2 * half;
            v2f a[2], b[4];
#pragma unroll
            for (int mt = 0; mt < 2; ++mt) {
                const int r = mw * 32 + mt * 16 + lr;
                a[mt].x = sA[cur][r][ka];
                a[mt].y = sA[cur][r][ka + 1];
            }
#pragma unroll
            for (int nt = 0; nt < 4; ++nt) {
                const int cx = nw * 64 + nt * 16 + lr;
                b[nt].x = sB[cur][ka][cx];
                b[nt].y = sB[cur][ka + 1][cx];
            }
#pragma unroll
            for (int mt = 0; mt < 2; ++mt)
#pragma unroll
                for (int nt = 0; nt < 4; ++nt)
                    acc[mt][nt] = wmma4(a[mt], b[nt], acc[mt][nt]);
        }
        __syncthreads();                  // reads done before buf reuse
    }

#pragma unroll
    for (int mt = 0; mt < 2; ++mt)
#pragma unroll
        for (int nt = 0; nt < 4; ++nt)
#pragma unroll
            for (int i = 0; i < 8; ++i) {
                const int r = m0 + mw * 32 + mt * 16 + i + 8 * half;
                const int cx = n0 + nw * 64 + nt * 16 + lr;
                C[(long)r * N + cx] = acc[mt][nt][i];
            }
}

// ---------------------------------------------------------------------------
// aug = concat(x, event_emb)  -> ROWS x 1040
// ---------------------------------------------------------------------------
__global__ __launch_bounds__(256)
void concat_kernel(const float* __restrict__ x, const float* __restrict__ ev,
                   float* __restrict__ aug) {
    const long idx = (long)blockIdx.x * blockDim.x + threadIdx.x;
    if (idx >= (long)ROWS * D_AUG) return;
    const int k = (int)(idx % D_AUG);
    const long m = idx / D_AUG;
    aug[idx] = (k < D_MODEL) ? x[m * D_MODEL + k] : ev[m * EVENT_DIM + (k - D_MODEL)];
}

// ---------------------------------------------------------------------------
// Causal depthwise conv1d (kernel 4) + bias + SiLU.  x_path = xz[:, :2048]
// ---------------------------------------------------------------------------
__global__ __launch_bounds__(256)
void conv_silu_kernel(const float* __restrict__ xz, const float* __restrict__ w,
                      const float* __restrict__ bias, float* __restrict__ xc) {
    const long idx = (long)blockIdx.x * blockDim.x + threadIdx.x;
    if (idx >= (long)ROWS * D_INNER) return;
    const int ch = (int)(idx & (D_INNER - 1));
    const int m  = (int)(idx >> 11);
    const int l  = m & (SEQ_L - 1);
    const int bi = m >> 12;
    float acc = bias[ch];
#pragma unroll
    for (int j = 0; j < D_CONV; ++j) {
        const int ls = l - (D_CONV - 1) + j;
        if (ls >= 0)
            acc += xz[(long)(bi * SEQ_L + ls) * (2 * D_INNER) + ch] * w[ch * D_CONV + j];
    }
    xc[idx] = siluf(acc);
}

// ---------------------------------------------------------------------------
// dt = softplus(xc @ dt_w + dt_b);  dA = dt * (-exp(A_log))
// ---------------------------------------------------------------------------
__global__ __launch_bounds__(256)
void dtA_kernel(const float* __restrict__ xc, const float* __restrict__ Wdt,
                const float* __restrict__ bdt, const float* __restrict__ A_log,
                float* __restrict__ dA) {
    const long idx = (long)blockIdx.x * blockDim.x + threadIdx.x;
    if (idx >= (long)ROWS * N_HEADS) return;
    const int h = (int)(idx & (N_HEADS - 1));
    const int m = (int)(idx >> 5);
    const float* row = xc + (long)m * D_INNER;
    float acc = bdt[h];
    for (int k = 0; k < D_INNER; ++k)
        acc += row[k] * Wdt[(long)k * N_HEADS + h];
    dA[idx] = -__expf(A_log[h]) * softplusf(acc);
}

// ---------------------------------------------------------------------------
// SSD chunked scan. One workgroup (4 waves) per (batch, head), sequential over
// 256 chunks. State S[64][64] in LDS; chunk tiles staged by double-buffered
// TDM; all matmuls fp32 WMMA. Fused epilogue: Y *= silu(z).
// ---------------------------------------------------------------------------
__global__ __launch_bounds__(128)
void ssd_scan_kernel(const float* __restrict__ xc, const float* __restrict__ Bs,
                     const float* __restrict__ Cs, const float* __restrict__ dA,
                     const float* __restrict__ xz, float* __restrict__ Ymul) {
    const int h  = blockIdx.x;
    const int bi = blockIdx.y;
    const int tid  = threadIdx.x;
    const int wave = tid >> 5;
    const int lane = tid & 31;
    const int half = lane >> 4;
    const int lr   = lane & 15;

    __shared__ float sX[2][CHUNK][HEAD_DIM];
    __shared__ float sB[2][CHUNK][D_STATE];
    __shared__ float sC[2][CHUNK][D_STATE];
    __shared__ float sS[D_STATE][HEAD_DIM];     // carried state
    __shared__ float sG[CHUNK][CHUNK];          // masked C*B^T
    __shared__ float sAcs[CHUNK], sDs[CHUNK], sSdo[CHUNK];
    __shared__ float sAtotExp;

    for (int i = tid; i < D_STATE * HEAD_DIM; i += 128)
        (&sS[0][0])[i] = 0.0f;

    const long gbase = (long)(bi * SEQ_L) * D_INNER + h * HEAD_DIM;
    if (tid < 32) {                              // prefetch chunk 0 -> buf 0
        tdm_load_2d(lds_addr_of(&sX[0][0][0]), xc + gbase, HEAD_DIM, CHUNK, D_INNER);
        tdm_load_2d(lds_addr_of(&sB[0][0][0]), Bs + gbase, HEAD_DIM, CHUNK, D_INNER);
        tdm_load_2d(lds_addr_of(&sC[0][0][0]), Cs + gbase, HEAD_DIM, CHUNK, D_INNER);
    }
    __syncthreads();

    for (int c = 0; c < NCHUNK; ++c) {
        const int cur = c & 1;
        const int lbase = c * CHUNK;
        if (tid < 32) {
            if (c + 1 < NCHUNK) {
                const int nxt = (c + 1) & 1;
                const long gn = gbase + (long)(lbase + CHUNK) * D_INNER;
                tdm_load_2d(lds_addr_of(&sX[nxt][0][0]), xc + gn, HEAD_DIM, CHUNK, D_INNER);
                tdm_load_2d(lds_addr_of(&sB[nxt][0][0]), Bs + gn, HEAD_DIM, CHUNK, D_INNER);
                tdm_load_2d(lds_addr_of(&sC[nxt][0][0]), Cs + gn, HEAD_DIM, CHUNK, D_INNER);
                __builtin_amdgcn_s_wait_tensorcnt(3);    // chunk c staged
            } else {
                __builtin_amdgcn_s_wait_tensorcnt(0);
            }
        }
        if (tid == 0) {
            float cs = 0.0f;
            for (int l = 0; l < CHUNK; ++l) {
                cs += dA[(long)(bi * SEQ_L + lbase + l) * N_HEADS + h];
                sAcs[l] = cs;
            }
            const float atot = cs;
            for (int l = 0; l < CHUNK; ++l) {
                sDs[l]  = __expf(atot - sAcs[l]);   // decay_states
                sSdo[l] = __expf(sAcs[l]);          // state_decay_out
            }
            sAtotExp = __expf(atot);
        }
        __syncthreads();

        // ---- G = C @ B^T (16x16, K=64), causal decay mask (wave 0) ----
        if (wave == 0) {
            v8f g = {};
            for (int k0 = 0; k0 < D_STATE; k0 += 4) {
                const int ka = k0 + 2 * half;
                v2f a; a.x = sC[cur][lr][ka]; a.y = sC[cur][lr][ka + 1];
                v2f b; b.x = sB[cur][lr][ka]; b.y = sB[cur][lr][ka + 1];
                g = wmma4(a, b, g);
            }
#pragma unroll
            for (int i = 0; i < 8; ++i) {
                const int r = i + 8 * half, s2 = lr;
                sG[r][s2] = (s2 <= r) ? g[i] * __expf(sAcs[r] - sAcs[s2]) : 0.0f;
            }
        }
        __syncthreads();

        // ---- Y = G @ X + diag(sdo) * C @ S ; wave owns p-tile 'wave' ----
        {
            const int pw = wave * 16;
            v8f y = {};
            for (int k0 = 0; k0 < CHUNK; k0 += 4) {          // Y_diag
                const int ka = k0 + 2 * half;
                v2f a; a.x = sG[lr][ka];            a.y = sG[lr][ka + 1];
                v2f b; b.x = sX[cur][ka][pw + lr];  b.y = sX[cur][ka + 1][pw + lr];
                y = wmma4(a, b, y);
            }
            const float sc = sSdo[lr];
            for (int k0 = 0; k0 < D_STATE; k0 += 4) {        // Y_off (carried S)
                const int ka = k0 + 2 * half;
                v2f a; a.x = sc * sC[cur][lr][ka];  a.y = sc * sC[cur][lr][ka + 1];
                v2f b; b.x = sS[ka][pw + lr];       b.y = sS[ka + 1][pw + lr];
                y = wmma4(a, b, y);
            }
#pragma unroll
            for (int i = 0; i < 8; ++i) {                     // fused * silu(z)
                const int  r = i + 8 * half;
                const long m = (long)(bi * SEQ_L + lbase + r);
                const int  p = pw + lr;
                const float zv = xz[m * (2 * D_INNER) + D_INNER + h * HEAD_DIM + p];
                Ymul[m * D_INNER + h * HEAD_DIM + p] = y[i] * siluf(zv);
            }
        }
        __syncthreads();

        // ---- S = exp(Atot)*S + B^T @ (diag(decay) * X) ; wave owns n-tile ----
        {
            const int n0  = wave * 16;
            const float eat = sAtotExp;
            v8f ns[4];
#pragma unroll
            for (int pt = 0; pt < 4; ++pt) {
                v8f a2;
#pragma unroll
                for (int i = 0; i < 8; ++i)
                    a2[i] = eat * sS[n0 + i + 8 * half][pt * 16 + lr];
                for (int k0 = 0; k0 < CHUNK; k0 += 4) {
                    const int ka = k0 + 2 * half;
                    v2f a; a.x = sB[cur][ka][n0 + lr] * sDs[ka];
                           a.y = sB[cur][ka + 1][n0 + lr] * sDs[ka + 1];
                    v2f b; b.x = sX[cur][ka][pt * 16 + lr];
                           b.y = sX[cur][ka + 1][pt * 16 + lr];
                    a2 = wmma4(a, b, a2);
                }
                ns[pt] = a2;
            }
            __syncthreads();                 // all reads of old S complete
#pragma unroll
            for (int pt = 0; pt < 4; ++pt)
#pragma unroll
                for (int i = 0; i < 8; ++i)
                    sS[n0 + i + 8 * half][pt * 16 + lr] = ns[pt][i];
        }
        __syncthreads();
    }
}

// ---------------------------------------------------------------------------
// Residual add + LayerNorm over d_model. One block per row.
// ---------------------------------------------------------------------------
__global__ __launch_bounds__(256)
void ln_kernel(const float* __restrict__ x, const float* __restrict__ outp,
               const float* __restrict__ gamma, const float* __restrict__ beta,
               float* __restrict__ out) {
    const int m   = blockIdx.x;
    const int tid = threadIdx.x;
    __shared__ float red[256];
    float v[4];
    float s = 0.0f;
#pragma unroll
    for (int i = 0; i < 4; ++i) {
        const int d = tid + i * 256;
        v[i] = x[(long)m * D_MODEL + d] + outp[(long)m * D_MODEL + d];
        s += v[i];
    }
    red[tid] = s; __syncthreads();
    for (int o = 128; o > 0; o >>= 1) {
        if (tid < o) red[tid] += red[tid + o];
        __syncthreads();
    }
    const float mu = red[0] * (1.0f / D_MODEL);
    __syncthreads();
    float sq = 0.0f;
#pragma unroll
    for (int i = 0; i < 4; ++i) { const float d = v[i] - mu; sq += d * d; }
    red[tid] = sq; __syncthreads();
    for (int o = 128; o > 0; o >>= 1) {
        if (tid < o) red[tid] += red[tid + o];
        __syncthreads();
    }
    const float inv = rsqrtf(red[0] * (1.0f / D_MODEL) + LN_EPS);
    __syncthreads();
#pragma unroll
    for (int i = 0; i < 4; ++i) {
        const int d = tid + i * 256;
        out[(long)m * D_MODEL + d] = (v[i] - mu) * inv * gamma[d] + beta[d];
    }
}

// ---------------------------------------------------------------------------
extern "C" void kernel_launch(void* const* d_in, const int* in_sizes, int n_in,
                              void* d_out, int out_size, void* d_ws, size_t ws_size,
                              hipStream_t stream) {
    const float* x     = (const float*)d_in[0];
    const float* ev    = (const float*)d_in[1];
    const float* inW   = (const float*)d_in[2];
    const float* convW = (const float*)d_in[3];
    const float* convB = (const float*)d_in[4];
    const float* A_log = (const float*)d_in[5];
    const float* BW    = (const float*)d_in[6];
    const float* CW    = (const float*)d_in[7];
    const float* dtW   = (const float*)d_in[8];
    const float* dtB   = (const float*)d_in[9];
    const float* outW  = (const float*)d_in[10];
    const float* gamma = (const float*)d_in[11];
    const float* beta  = (const float*)d_in[12];
    float* out = (float*)d_out;

    float* ws = (float*)d_ws;
    float* aug  = ws;                                  // ROWS * 1040
    float* xz   = aug  + (size_t)ROWS * D_AUG;         // ROWS * 4096
    float* xc   = xz   + (size_t)ROWS * 2 * D_INNER;   // ROWS * 2048
    float* dA   = xc   + (size_t)ROWS * D_INNER;       // ROWS * 32
    float* Bss  = dA   + (size_t)ROWS * N_HEADS;       // ROWS * 2048
    float* Css  = Bss  + (size_t)ROWS * D_INNER;       // ROWS * 2048
    float* Ymul = Css  + (size_t)ROWS * D_INNER;       // ROWS * 2048
    float* proj = Ymul + (size_t)ROWS * D_INNER;       // ROWS * 1024

    // 0. aug = [x | event_emb]
    concat_kernel<<<(int)(((long)ROWS * D_AUG + 255) / 256), 256, 0, stream>>>(x, ev, aug);
    // 1. in_proj GEMM (TDM + WMMA): xz = aug @ inW   (K=1040 = 65*16)
    gemm_tdm_wmma<<<dim3((2 * D_INNER) / BN, ROWS / BM), 256, 0, stream>>>(
        aug, inW, xz, ROWS, 2 * D_INNER, D_AUG);
    // 2. causal depthwise conv + SiLU
    conv_silu_kernel<<<(ROWS * D_INNER) / 256, 256, 0, stream>>>(xz, convW, convB, xc);
    // 3. dt -> softplus -> dA
    dtA_kernel<<<(ROWS * N_HEADS) / 256, 256, 0, stream>>>(xc, dtW, dtB, A_log, dA);
    // 4/5. B and C projections (TDM + WMMA)
    gemm_tdm_wmma<<<dim3(D_INNER / BN, ROWS / BM), 256, 0, stream>>>(
        xc, BW, Bss, ROWS, D_INNER, D_INNER);
    gemm_tdm_wmma<<<dim3(D_INNER / BN, ROWS / BM), 256, 0, stream>>>(
        xc, CW, Css, ROWS, D_INNER, D_INNER);
    // 6. chunked SSD scan (TDM + WMMA) fused with * silu(z)
    ssd_scan_kernel<<<dim3(N_HEADS, BATCH), 128, 0, stream>>>(xc, Bss, Css, dA, xz, Ymul);
    // 7. out_proj GEMM (TDM + WMMA)
    gemm_tdm_wmma<<<dim3(D_MODEL / BN, ROWS / BM), 256, 0, stream>>>(
        Ymul, outW, proj, ROWS, D_MODEL, D_INNER);
    // 8. residual + LayerNorm
    ln_kernel<<<ROWS, 256, 0, stream>>>(x, proj, gamma, beta, out);
}